// VQWeightedAvgPool_4398046511223
// MI455X (gfx1250) — compile-verified
//
#include <hip/hip_runtime.h>

typedef __attribute__((ext_vector_type(2))) float v2f;
typedef __attribute__((ext_vector_type(8))) float v8f;

#define MAX_L 2048
#define T1 256

// ---------------- Phase 1: per-batch segment weights -> d_ws ----------------
// w[b,l] = valid ? 1 / max(n_seg * seg_len[l], 1) : 0
__global__ void vq_weights_kernel(const long long* __restrict__ lengths,
                                  const long long* __restrict__ vq,
                                  float* __restrict__ w, int L) {
    const int b = blockIdx.x;
    const int t = threadIdx.x;

    __shared__ int seg_sm[MAX_L];     // local inclusive count, then seg_id
    __shared__ int counts_sm[MAX_L];  // per-segment valid count
    __shared__ int scan_sm[T1];
    __shared__ int nseg_sm;

    long long len64 = lengths[b];
    int len = (int)(len64 < 0 ? 0 : (len64 > (long long)L ? (long long)L : len64));
    const long long* vqb = vq + (long long)b * L * 2;

    for (int l = t; l < L; l += T1) counts_sm[l] = 0;

    const int chunk = (L + T1 - 1) / T1;
    const int l0 = t * chunk;
    const int l1 = (l0 + chunk < L) ? (l0 + chunk) : L;

    // local boundary pass
    int run = 0;
    for (int l = l0; l < l1; ++l) {
        bool valid = (l < len);
        bool diff;
        if (l == 0) {
            diff = true;
        } else {
            diff = (vqb[2 * l] != vqb[2 * (l - 1)]) ||
                   (vqb[2 * l + 1] != vqb[2 * (l - 1) + 1]);
        }
        run += (diff && valid) ? 1 : 0;
        seg_sm[l] = run;  // local inclusive boundary count
    }
    scan_sm[t] = run;
    __syncthreads();

    // Hillis-Steele inclusive scan across thread totals
    for (int off = 1; off < T1; off <<= 1) {
        int v = (t >= off) ? scan_sm[t - off] : 0;
        __syncthreads();
        scan_sm[t] += v;
        __syncthreads();
    }
    int excl = scan_sm[t] - run;
    if (t == T1 - 1) nseg_sm = scan_sm[T1 - 1];
    __syncthreads();

    // finalize seg_id + histogram of valid counts per segment
    for (int l = l0; l < l1; ++l) {
        int s = excl + seg_sm[l] - 1;
        if (s < 0) s = 0;
        seg_sm[l] = s;
        if (l < len) atomicAdd(&counts_sm[s], 1);
    }
    __syncthreads();

    float nseg = (float)nseg_sm;
    float* wb = w + (size_t)b * L;
    for (int l = t; l < L; l += T1) {
        float wv = 0.0f;
        if (l < len) {
            float denom = nseg * (float)counts_sm[seg_sm[l]];
            denom = fmaxf(denom, 1.0f);
            wv = 1.0f / denom;
        }
        wb[l] = wv;
    }
}

// ---------------- Phase 2: out[b,:] = sum_l w[b,l] * x[b,l,:] via WMMA ------
// Specialized for D=768 so every B load in the unrolled body has a
// compile-time immediate offset (single loop-carried pointer bump).
//
// Each wave owns a 16-wide d tile. A (16x4 f32) = w chunk broadcast across all
// 16 M rows; B (4x16 f32) = x[l0..l0+3, d0..d0+15]. f32 K layout: lane-half
// g = lane>>4, VGPR v -> K = v + 2g. Two accumulators break the
// WMMA->WMMA RAW chain through C.
__global__ void __launch_bounds__(256)
vq_wmma_pool_768(const float* __restrict__ x,   // pre-offset to [:, N-1]
                 const float* __restrict__ w,
                 float* __restrict__ out,
                 int L, long long batch_stride) {
    constexpr int D = 768;
    const int b = blockIdx.y;

    __shared__ float w_sm[MAX_L];
    const float* wb = w + (size_t)b * L;
    for (int l = threadIdx.x; l < L; l += blockDim.x) w_sm[l] = wb[l];
    __syncthreads();

    const int wave = threadIdx.x >> 5;
    const int lane = threadIdx.x & 31;
    const int d0 = (blockIdx.x * 8 + wave) * 16;   // 48 tiles, 8 waves/block

    const int g = lane >> 4;   // lane half selects K pair
    const int n = lane & 15;   // column within d tile

    const float* p = x + (size_t)b * batch_stride + (size_t)d0 + n
                       + (size_t)(2 * g) * D;
    const v2f* w2g = (const v2f*)w_sm + g;

    v8f c0 = {0.f, 0.f, 0.f, 0.f, 0.f, 0.f, 0.f, 0.f};
    v8f c1 = {0.f, 0.f, 0.f, 0.f, 0.f, 0.f, 0.f, 0.f};

    // 32 l-values (8 WMMAs, 16 NT loads, 4 ds_load_2addr_b64) per body.
    for (int l0 = 0; l0 < L; l0 += 32) {
#pragma unroll
        for (int j = 0; j < 8; ++j) {
            v2f a = w2g[(l0 >> 1) + 2 * j];           // w[l0+4j+2g], w[+1]
            v2f bm;
            bm.x = __builtin_nontemporal_load(p + (4 * j) * D);      // imm off
            bm.y = __builtin_nontemporal_load(p + (4 * j + 1) * D);  // imm off
            if (j & 1)
                c1 = __builtin_amdgcn_wmma_f32_16x16x4_f32(
                        false, a, false, bm, (short)0, c1, false, false);
            else
                c0 = __builtin_amdgcn_wmma_f32_16x16x4_f32(
                        false, a, false, bm, (short)0, c0, false, false);
        }
        p += 32 * (size_t)D;  // single loop-carried pointer bump
    }

    v8f c = c0 + c1;
    // All 16 C rows are identical (A rows broadcast); row M=0 lives in
    // VGPR0 of lanes 0..15.
    if (lane < 16) out[(size_t)b * D + d0 + lane] = c[0];
}

// ---------------- Fallback for unexpected shapes (plain VALU) ---------------
__global__ void vq_pool_fallback(const float* __restrict__ x,
                                 const float* __restrict__ w,
                                 float* __restrict__ out,
                                 int L, int D, long long batch_stride) {
    const int b = blockIdx.y;
    const int d = blockIdx.x * blockDim.x + threadIdx.x;
    if (d >= D) return;
    const float* xb = x + (size_t)b * batch_stride + d;
    const float* wb = w + (size_t)b * L;
    float acc = 0.f;
    for (int l = 0; l < L; ++l) acc += wb[l] * xb[(size_t)l * D];
    out[(size_t)b * D + d] = acc;
}

// ---------------- Host launch ----------------
extern "C" void kernel_launch(void* const* d_in, const int* in_sizes, int n_in,
                              void* d_out, int out_size, void* d_ws, size_t ws_size,
                              hipStream_t stream) {
    const float* feat = (const float*)d_in[0];
    const long long* lengths = (const long long*)d_in[1];
    const long long* vq = (const long long*)d_in[2];
    float* out = (float*)d_out;

    const int B = in_sizes[1];                          // 16
    const int L = (int)(in_sizes[2] / (B * 2));         // 2048
    const int D = out_size / B;                         // 768
    const long long feat_elems = (long long)in_sizes[0];
    const int N = (int)(feat_elems / ((long long)B * L * D));  // 13

    float* w = (float*)d_ws;  // B*L floats (128 KB)

    vq_weights_kernel<<<B, T1, 0, stream>>>(lengths, vq, w, L);

    const float* x_last = feat + (long long)(N - 1) * L * D;   // [:, -1]
    const long long bstride = (long long)N * L * D;

    if (D == 768 && (L % 32) == 0 && L <= MAX_L) {
        dim3 grid2(6, B);  // 48 d-tiles / 8 waves per block
        vq_wmma_pool_768<<<grid2, 256, 0, stream>>>(x_last, w, out, L, bstride);
    } else {
        dim3 grid2((D + 255) / 256, B);
        vq_pool_fallback<<<grid2, 256, 0, stream>>>(x_last, w, out, L, D, bstride);
    }
}